// GraphSAGE_51591147159576
// MI455X (gfx1250) — compile-verified
//
#include <hip/hip_runtime.h>
#include <math.h>

typedef __attribute__((ext_vector_type(2))) float v2f;
typedef __attribute__((ext_vector_type(8))) float v8f;

#define C_DIM 256
#define OUT_C 10

// ---------------- utility ----------------
__global__ void zero_i32(int* __restrict__ p, int n) {
    int i = blockIdx.x * blockDim.x + threadIdx.x;
    if (i < n) p[i] = 0;
}

// ---------------- CSR build ----------------
__global__ void edge_hist(const int* __restrict__ dst, int e, int* __restrict__ deg) {
    int i = blockIdx.x * blockDim.x + threadIdx.x;
    if (i < e) atomicAdd(&deg[dst[i]], 1);
}

__global__ __launch_bounds__(256) void scan_blocks(const int* __restrict__ deg, int n,
                                                   int* __restrict__ excl,  // block-local exclusive
                                                   int* __restrict__ bsum) {
    __shared__ int sh[256];
    int gid = blockIdx.x * 256 + threadIdx.x;
    int v = (gid < n) ? deg[gid] : 0;
    sh[threadIdx.x] = v;
    __syncthreads();
    for (int off = 1; off < 256; off <<= 1) {
        int t = (threadIdx.x >= off) ? sh[threadIdx.x - off] : 0;
        __syncthreads();
        sh[threadIdx.x] += t;
        __syncthreads();
    }
    if (gid < n) excl[gid] = sh[threadIdx.x] - v;
    if (threadIdx.x == 255) bsum[blockIdx.x] = sh[255];
}

__global__ __launch_bounds__(256) void scan_sums(int* __restrict__ bsum, int nb) {
    __shared__ int sh[256];
    int v = (threadIdx.x < nb) ? bsum[threadIdx.x] : 0;
    sh[threadIdx.x] = v;
    __syncthreads();
    for (int off = 1; off < 256; off <<= 1) {
        int t = (threadIdx.x >= off) ? sh[threadIdx.x - off] : 0;
        __syncthreads();
        sh[threadIdx.x] += t;
        __syncthreads();
    }
    if (threadIdx.x < nb) bsum[threadIdx.x] = sh[threadIdx.x] - v;  // exclusive
}

__global__ void add_offsets(int* __restrict__ row_ptr, const int* __restrict__ bsum,
                            int n, int total, int* __restrict__ cursor) {
    int gid = blockIdx.x * blockDim.x + threadIdx.x;
    if (gid < n) {
        int v = row_ptr[gid] + bsum[gid >> 8];
        row_ptr[gid] = v;
        cursor[gid]  = v;
    }
    if (gid == 0) row_ptr[n] = total;
}

__global__ void edge_fill(const int* __restrict__ src, const int* __restrict__ dst, int e,
                          int* __restrict__ cursor, int* __restrict__ col) {
    int i = blockIdx.x * blockDim.x + threadIdx.x;
    if (i < e) {
        int p = atomicAdd(&cursor[dst[i]], 1);
        col[p] = src[i];
    }
}

// ---------------- mean aggregation (gather, no float atomics) ----------------
// block = 256 threads (one channel each), one block per node; feature rows are
// L2-resident (each feat buffer is 51.2MB, x+agg+h ~ 153MB < 192MB global L2).
__global__ __launch_bounds__(256) void gather_mean(const float* __restrict__ feat,
                                                   const int* __restrict__ row_ptr,
                                                   const int* __restrict__ col,
                                                   float* __restrict__ out) {
    int node = blockIdx.x;
    int c    = threadIdx.x;
    int s = row_ptr[node];
    int e = row_ptr[node + 1];
    float sum = 0.0f;
    for (int j = s; j < e; ++j) {
        int srcn = col[j];
        if (j + 1 < e) __builtin_prefetch(&feat[(size_t)col[j + 1] * C_DIM + c], 0, 1);
        sum += feat[(size_t)srcn * C_DIM + c];
    }
    float d = (float)(e - s);
    out[(size_t)node * C_DIM + c] = sum / fmaxf(d, 1.0f);
}

// ---------------- layer 1 GEMM: h = relu(agg@W1l + x@W1r + b1) ----------------
// One wave computes a 32(M) x 64(N) tile with V_WMMA_F32_16X16X4_F32 so the
// (shared-across-M) B loads are amortized: per K-step 16 WMMAs vs 8 A-dwords +
// 16 B-dwords of loads. Fragment layouts per ISA 7.12.2:
//   A 16x4: lanes 0-15 hold K={k,k+1}, lanes 16-31 hold K={k+2,k+3}, M=lane&15
//   B 4x16 / C,D 16x16: row-striped across lanes within a VGPR.
__global__ __launch_bounds__(32) void sage_gemm1(const float* __restrict__ agg,
                                                 const float* __restrict__ x,
                                                 const float* __restrict__ W1l,
                                                 const float* __restrict__ W1r,
                                                 const float* __restrict__ b1,
                                                 float* __restrict__ h, int n_nodes) {
    const int m0   = blockIdx.x * 32;  // two 16-row tiles per wave
    const int n0   = blockIdx.y * 64;
    const int lane = threadIdx.x;
    const int r    = lane & 15;
    const int hi   = lane >> 4;  // 0 or 1

    int arow0 = m0 + r;
    int arow1 = m0 + 16 + r;
    int arow0c = arow0 < n_nodes ? arow0 : (n_nodes - 1);
    int arow1c = arow1 < n_nodes ? arow1 : (n_nodes - 1);

    v8f c[2][4];
#pragma unroll
    for (int mt = 0; mt < 2; ++mt)
#pragma unroll
        for (int t = 0; t < 4; ++t) c[mt][t] = (v8f){};

    for (int k = 0; k < C_DIM; k += 4) {
        const int ka = k + (hi << 1);  // k or k+2
        v2f aA0 = *(const v2f*)(agg + (size_t)arow0c * C_DIM + ka);
        v2f aX0 = *(const v2f*)(x   + (size_t)arow0c * C_DIM + ka);
        v2f aA1 = *(const v2f*)(agg + (size_t)arow1c * C_DIM + ka);
        v2f aX1 = *(const v2f*)(x   + (size_t)arow1c * C_DIM + ka);
#pragma unroll
        for (int t = 0; t < 4; ++t) {
            const int n = n0 + t * 16 + r;
            v2f bl, br;
            bl.x = W1l[(size_t)ka * C_DIM + n];
            bl.y = W1l[(size_t)(ka + 1) * C_DIM + n];
            br.x = W1r[(size_t)ka * C_DIM + n];
            br.y = W1r[(size_t)(ka + 1) * C_DIM + n];
            c[0][t] = __builtin_amdgcn_wmma_f32_16x16x4_f32(false, aA0, false, bl,
                                                            (short)0, c[0][t], false, false);
            c[0][t] = __builtin_amdgcn_wmma_f32_16x16x4_f32(false, aX0, false, br,
                                                            (short)0, c[0][t], false, false);
            c[1][t] = __builtin_amdgcn_wmma_f32_16x16x4_f32(false, aA1, false, bl,
                                                            (short)0, c[1][t], false, false);
            c[1][t] = __builtin_amdgcn_wmma_f32_16x16x4_f32(false, aX1, false, br,
                                                            (short)0, c[1][t], false, false);
        }
    }

    // D layout: VGPR i of tile (mt,t) -> row m0 + mt*16 + i + hi*8, col n0 + t*16 + r
    const bool full_tile = (m0 + 32 <= n_nodes);
    if (full_tile) {
#pragma unroll
        for (int mt = 0; mt < 2; ++mt)
#pragma unroll
            for (int t = 0; t < 4; ++t) {
                const int n     = n0 + t * 16 + r;
                const float bias = b1[n];
#pragma unroll
                for (int i = 0; i < 8; ++i) {
                    int row = m0 + mt * 16 + i + hi * 8;
                    float v = c[mt][t][i] + bias;
                    h[(size_t)row * C_DIM + n] = v > 0.0f ? v : 0.0f;
                }
            }
    } else {
#pragma unroll
        for (int mt = 0; mt < 2; ++mt)
#pragma unroll
            for (int t = 0; t < 4; ++t) {
                const int n     = n0 + t * 16 + r;
                const float bias = b1[n];
#pragma unroll
                for (int i = 0; i < 8; ++i) {
                    int row = m0 + mt * 16 + i + hi * 8;
                    if (row < n_nodes) {
                        float v = c[mt][t][i] + bias;
                        h[(size_t)row * C_DIM + n] = v > 0.0f ? v : 0.0f;
                    }
                }
            }
    }
}

// ---------------- layer 2 + log_softmax ----------------
// One wave per node: lane-parallel over channels, shfl_xor reduction, fused
// bias + log_softmax; N=10 is far too small for WMMA to pay off.
__global__ __launch_bounds__(32) void sage_layer2(const float* __restrict__ agg,
                                                  const float* __restrict__ h,
                                                  const float* __restrict__ W2l,
                                                  const float* __restrict__ W2r,
                                                  const float* __restrict__ b2,
                                                  float* __restrict__ out) {
    const int node = blockIdx.x;
    const int lane = threadIdx.x;

    float a[8], hv[8];
#pragma unroll
    for (int k = 0; k < 8; ++k) {
        int ch = lane + 32 * k;
        a[k]  = agg[(size_t)node * C_DIM + ch];
        hv[k] = h[(size_t)node * C_DIM + ch];
    }

    float o[OUT_C];
#pragma unroll
    for (int j = 0; j < OUT_C; ++j) {
        float s = 0.0f;
#pragma unroll
        for (int k = 0; k < 8; ++k) {
            int ch = lane + 32 * k;
            s += a[k] * W2l[(size_t)ch * OUT_C + j] + hv[k] * W2r[(size_t)ch * OUT_C + j];
        }
#pragma unroll
        for (int off = 16; off > 0; off >>= 1) s += __shfl_xor(s, off, 32);
        o[j] = s + b2[j];  // identical in every lane after full xor-reduction
    }

    float m = o[0];
#pragma unroll
    for (int j = 1; j < OUT_C; ++j) m = fmaxf(m, o[j]);
    float lse = 0.0f;
#pragma unroll
    for (int j = 0; j < OUT_C; ++j) lse += expf(o[j] - m);
    lse = m + logf(lse);

    if (lane == 0) {
#pragma unroll
        for (int j = 0; j < OUT_C; ++j) out[(size_t)node * OUT_C + j] = o[j] - lse;
    }
}

// ---------------- launcher ----------------
static inline size_t align256(size_t v) { return (v + 255) & ~(size_t)255; }

extern "C" void kernel_launch(void* const* d_in, const int* in_sizes, int n_in,
                              void* d_out, int out_size, void* d_ws, size_t ws_size,
                              hipStream_t stream) {
    const float* x   = (const float*)d_in[0];
    const int*   ei  = (const int*)d_in[1];
    const float* W1l = (const float*)d_in[2];
    const float* W1r = (const float*)d_in[3];
    const float* b1  = (const float*)d_in[4];
    const float* W2l = (const float*)d_in[5];
    const float* W2r = (const float*)d_in[6];
    const float* b2  = (const float*)d_in[7];

    const int N = in_sizes[0] / C_DIM;  // 50000
    const int E = in_sizes[1] / 2;      // 1600000
    const int* src = ei;
    const int* dst = ei + E;

    // workspace carve-up
    char* w = (char*)d_ws;
    size_t off = 0;
    int* deg     = (int*)(w + off); off = align256(off + (size_t)N * 4);
    int* row_ptr = (int*)(w + off); off = align256(off + (size_t)(N + 1) * 4);
    int* cursor  = (int*)(w + off); off = align256(off + (size_t)N * 4);
    int* bsum    = (int*)(w + off); off = align256(off + 256 * 4);
    int* col     = (int*)(w + off); off = align256(off + (size_t)E * 4);
    float* agg   = (float*)(w + off); off = align256(off + (size_t)N * C_DIM * 4);
    float* hbuf  = (float*)(w + off); off = align256(off + (size_t)N * C_DIM * 4);
    (void)ws_size; (void)n_in; (void)out_size;

    const int nb = (N + 255) / 256;

    // 1) CSR build
    zero_i32<<<nb, 256, 0, stream>>>(deg, N);
    edge_hist<<<(E + 255) / 256, 256, 0, stream>>>(dst, E, deg);
    scan_blocks<<<nb, 256, 0, stream>>>(deg, N, row_ptr, bsum);
    scan_sums<<<1, 256, 0, stream>>>(bsum, nb);
    add_offsets<<<nb, 256, 0, stream>>>(row_ptr, bsum, N, E, cursor);
    edge_fill<<<(E + 255) / 256, 256, 0, stream>>>(src, dst, E, cursor, col);

    // 2) layer 1: mean-aggregate x, fused WMMA GEMM + bias + ReLU
    gather_mean<<<N, 256, 0, stream>>>(x, row_ptr, col, agg);
    sage_gemm1<<<dim3((N + 31) / 32, C_DIM / 64), 32, 0, stream>>>(agg, x, W1l, W1r, b1,
                                                                   hbuf, N);

    // 3) layer 2: mean-aggregate h, small fused GEMM + log_softmax
    gather_mean<<<N, 256, 0, stream>>>(hbuf, row_ptr, col, agg);
    sage_layer2<<<N, 32, 0, stream>>>(agg, hbuf, W2l, W2r, b2, (float*)d_out);
}